// GraphAttentionLayer_24318104830717
// MI455X (gfx1250) — compile-verified
//
#include <hip/hip_runtime.h>

typedef __attribute__((ext_vector_type(2))) float v2f;
typedef __attribute__((ext_vector_type(8))) float v8f;

#define NN 400      // nodes
#define DI 256      // d_in
#define DO 128      // d_out
#define ALPHA 0.2f
#define NEG_BIG -9e15f

#define GI 20       // i-groups (each 20 i's)
#define GJ 20       // j-groups (each 20 j's)
#define TI (NN / GI)  // 20 i per group
#define TJ (NN / GJ)  // 20 j per group

// ---------------------------------------------------------------------------
// WMMA f32 16x16 tile accumulate: D += A[MxK] * B[KxN]   (row-major inputs)
// A 16x4 f32 layout : lane L (half=L/16, lr=L%16): v0 = A[m0+lr][k+2*half],
//                     v1 = A[m0+lr][k+2*half+1]
// B 4x16 f32 layout : v0 = B[k+2*half][n0+lr], v1 = B[k+2*half+1][n0+lr]
// C/D 16x16 layout  : vgpr v, lane L -> D[m0 + v + 8*half][n0 + lr]
// ---------------------------------------------------------------------------
__device__ __forceinline__ v8f wmma_acc_tile(const float* __restrict__ A, int lda,
                                             const float* __restrict__ B, int ldb,
                                             int m0, int n0, int K, v8f acc) {
  const int lane = threadIdx.x & 31;
  const int half = lane >> 4;
  const int lr = lane & 15;
  const float* arow = A + (size_t)(m0 + lr) * lda;
  for (int k0 = 0; k0 < K; k0 += 4) {
    const int ka = k0 + 2 * half;
    v2f a, b;
    a.x = arow[ka];
    a.y = arow[ka + 1];
    b.x = B[(size_t)ka * ldb + n0 + lr];
    b.y = B[(size_t)(ka + 1) * ldb + n0 + lr];
    acc = __builtin_amdgcn_wmma_f32_16x16x4_f32(false, a, false, b, (short)0, acc,
                                                false, false);
  }
  return acc;
}

// Same, but A is given transposed in memory: At[k][m] (ldat = stride of k rows)
__device__ __forceinline__ v8f wmma_acc_tile_at(const float* __restrict__ At, int ldat,
                                                const float* __restrict__ B, int ldb,
                                                int m0, int n0, int K, v8f acc) {
  const int lane = threadIdx.x & 31;
  const int half = lane >> 4;
  const int lr = lane & 15;
  for (int k0 = 0; k0 < K; k0 += 4) {
    const int ka = k0 + 2 * half;
    v2f a, b;
    a.x = At[(size_t)ka * ldat + m0 + lr];
    a.y = At[(size_t)(ka + 1) * ldat + m0 + lr];
    b.x = B[(size_t)ka * ldb + n0 + lr];
    b.y = B[(size_t)(ka + 1) * ldb + n0 + lr];
    acc = __builtin_amdgcn_wmma_f32_16x16x4_f32(false, a, false, b, (short)0, acc,
                                                false, false);
  }
  return acc;
}

__device__ __forceinline__ void store_tile(float* __restrict__ D, int ldd,
                                           int m0, int n0, v8f acc) {
  const int lane = threadIdx.x & 31;
  const int half = lane >> 4;
  const int lr = lane & 15;
#pragma unroll
  for (int v = 0; v < 8; ++v)
    D[(size_t)(m0 + v + 8 * half) * ldd + n0 + lr] = acc[v];
}

// ---- Kernel 1: wh = h @ W   [400x256]@[256x128], 25x8 tiles ---------------
__global__ void k_gemm_wh(const float* __restrict__ h, const float* __restrict__ W,
                          float* __restrict__ wh) {
  const int mt = blockIdx.x / 8, nt = blockIdx.x % 8;
  v8f acc = {};
  acc = wmma_acc_tile(h, DI, W, DO, mt * 16, nt * 16, DI, acc);
  store_tile(wh, DO, mt * 16, nt * 16, acc);
}

// ---- Kernel 2: S1 = wh @ a1[0:128,:], S2 = wh @ a1[128:256,:] -------------
__global__ void k_gemm_s(const float* __restrict__ wh, const float* __restrict__ a1,
                         float* __restrict__ S1, float* __restrict__ S2) {
  const int which = blockIdx.x / 625;
  const int t = blockIdx.x % 625;
  const int m0 = (t / 25) * 16, n0 = (t % 25) * 16;
  const float* B = a1 + (which ? (size_t)128 * NN : 0);
  float* D = which ? S2 : S1;
  v8f acc = {};
  acc = wmma_acc_tile(wh, DO, B, NN, m0, n0, DO, acc);
  store_tile(D, NN, m0, n0, acc);
}

// ---- Kernel 3: SINGLE non-temporal pass over a3 (256 MB) ------------------
// Grid: 400 blocks = (g in [0,20)) x (f in [0,20)); block owns i in
// [g*20,g*20+20), j in [f*20,f*20+20), all 400 r columns.
//   B1part[f][i][r] = sum_{j in f} a3[i*400+j][r]      (reduce over f later)
//   B2part[g][j][r] = sum_{i in g} a3[i*400+j][r]      (reduce over g later)
//   Cpart[blk][r*5+e] = sum_{tile} a3[p][r] * ef[p][e] (reduce over blk later)
__global__ void k_reduce_a3(const float* __restrict__ a3, const float* __restrict__ ef,
                            float* __restrict__ B1part, float* __restrict__ B2part,
                            float* __restrict__ Cpart) {
  const int g = blockIdx.x / GJ;
  const int f = blockIdx.x % GJ;
  const int tid = threadIdx.x;         // 256 threads
  const int r0 = tid, r1 = tid + 256;  // each thread owns up to 2 r-columns
  const bool has1 = (r1 < NN);

  float b1a[TI], b1b[TI];
#pragma unroll
  for (int ii = 0; ii < TI; ++ii) { b1a[ii] = 0.f; b1b[ii] = 0.f; }
  float c0[5] = {0.f, 0.f, 0.f, 0.f, 0.f};
  float c1[5] = {0.f, 0.f, 0.f, 0.f, 0.f};

  for (int jj = 0; jj < TJ; ++jj) {
    const int j = f * TJ + jj;
    float b2a = 0.f, b2b = 0.f;
#pragma unroll
    for (int ii = 0; ii < TI; ++ii) {
      const int i = g * TI + ii;
      const int p = i * NN + j;
      const float* row = a3 + (size_t)p * NN;
      float e[5];
#pragma unroll
      for (int k = 0; k < 5; ++k) e[k] = ef[(size_t)p * 5 + k];
      const float v0 = __builtin_nontemporal_load(row + r0);
      b1a[ii] += v0;
      b2a += v0;
#pragma unroll
      for (int k = 0; k < 5; ++k) c0[k] += v0 * e[k];
      if (has1) {
        const float v1 = __builtin_nontemporal_load(row + r1);
        b1b[ii] += v1;
        b2b += v1;
#pragma unroll
        for (int k = 0; k < 5; ++k) c1[k] += v1 * e[k];
      }
    }
    float* b2row = B2part + ((size_t)g * NN + j) * NN;
    b2row[r0] = b2a;
    if (has1) b2row[r1] = b2b;
  }
#pragma unroll
  for (int ii = 0; ii < TI; ++ii) {
    const int i = g * TI + ii;
    float* b1row = B1part + ((size_t)f * NN + i) * NN;
    b1row[r0] = b1a[ii];
    if (has1) b1row[r1] = b1b[ii];
  }
  float* crow = Cpart + (size_t)blockIdx.x * (NN * 5);
#pragma unroll
  for (int k = 0; k < 5; ++k) crow[r0 * 5 + k] = c0[k];
  if (has1) {
#pragma unroll
    for (int k = 0; k < 5; ++k) crow[r1 * 5 + k] = c1[k];
  }
}

// ---- Kernel 3b: fold partials: B1[i][r], B2[j][r], C[r*5+e] ---------------
__global__ void k_reduce_parts(const float* __restrict__ B1part,
                               const float* __restrict__ B2part,
                               const float* __restrict__ Cpart,
                               float* __restrict__ B1, float* __restrict__ B2,
                               float* __restrict__ C) {
  const int idx = blockIdx.x * 256 + threadIdx.x;
  if (idx < NN * NN) {
    float s = 0.f;
#pragma unroll
    for (int fset = 0; fset < GJ; ++fset) s += B1part[(size_t)fset * (NN * NN) + idx];
    B1[idx] = s;
  } else if (idx < 2 * NN * NN) {
    const int t = idx - NN * NN;
    float s = 0.f;
#pragma unroll
    for (int gset = 0; gset < GI; ++gset) s += B2part[(size_t)gset * (NN * NN) + t];
    B2[t] = s;
  } else if (idx < 2 * NN * NN + NN * 5) {
    const int t = idx - 2 * NN * NN;
    float s = 0.f;
    for (int b = 0; b < GI * GJ; ++b) s += Cpart[(size_t)b * (NN * 5) + t];
    C[t] = s;
  }
}

// ---- Kernel 4: e = leaky(B1^T@S1 + B2^T@S2 + C@a1e), diag -> NEG_BIG ------
// B1 is stored [i][r] (so A = B1^T is read via the transposed-A helper).
__global__ void k_gemm_e(const float* __restrict__ B1, const float* __restrict__ S1,
                         const float* __restrict__ B2, const float* __restrict__ S2,
                         const float* __restrict__ C, const float* __restrict__ a1,
                         float* __restrict__ ebuf) {
  const int mt = blockIdx.x / 25, nt = blockIdx.x % 25;
  const int m0 = mt * 16, n0 = nt * 16;
  v8f acc = {};
  acc = wmma_acc_tile_at(B1, NN, S1, NN, m0, n0, NN, acc);
  acc = wmma_acc_tile_at(B2, NN, S2, NN, m0, n0, NN, acc);
  const int lane = threadIdx.x & 31;
  const int half = lane >> 4, lr = lane & 15;
  const float* a1e = a1 + (size_t)256 * NN;  // a1 rows 256..260
#pragma unroll
  for (int v = 0; v < 8; ++v) {
    const int m = m0 + v + 8 * half;
    const int q = n0 + lr;
    float x = acc[v];
#pragma unroll
    for (int e = 0; e < 5; ++e) x += C[m * 5 + e] * a1e[(size_t)e * NN + q];
    x = (x > 0.f) ? x : ALPHA * x;
    if (m == q) x = NEG_BIG;
    ebuf[(size_t)m * NN + q] = x;
  }
}

// ---- Kernel 5: row softmax over [400x400] ---------------------------------
__global__ void k_softmax(float* __restrict__ ebuf) {
  __shared__ float sred[256];
  const int r = blockIdx.x, tid = threadIdx.x;
  float* row = ebuf + (size_t)r * NN;
  float mx = -3.4e38f;
  for (int c = tid; c < NN; c += 256) mx = fmaxf(mx, row[c]);
  sred[tid] = mx;
  __syncthreads();
  for (int s = 128; s > 0; s >>= 1) {
    if (tid < s) sred[tid] = fmaxf(sred[tid], sred[tid + s]);
    __syncthreads();
  }
  mx = sred[0];
  __syncthreads();
  float sum = 0.f;
  for (int c = tid; c < NN; c += 256) {
    const float ev = __expf(row[c] - mx);
    row[c] = ev;
    sum += ev;
  }
  sred[tid] = sum;
  __syncthreads();
  for (int s = 128; s > 0; s >>= 1) {
    if (tid < s) sred[tid] += sred[tid + s];
    __syncthreads();
  }
  const float inv = 1.f / sred[0];
  for (int c = tid; c < NN; c += 256) row[c] *= inv;
}

// ---- Kernel 6: h' = attention @ wh  [400x400]@[400x128] -------------------
__global__ void k_gemm_out(const float* __restrict__ att, const float* __restrict__ wh,
                           float* __restrict__ out) {
  const int mt = blockIdx.x / 8, nt = blockIdx.x % 8;
  v8f acc = {};
  acc = wmma_acc_tile(att, NN, wh, DO, mt * 16, nt * 16, NN, acc);
  store_tile(out, DO, mt * 16, nt * 16, acc);
}

extern "C" void kernel_launch(void* const* d_in, const int* in_sizes, int n_in,
                              void* d_out, int out_size, void* d_ws, size_t ws_size,
                              hipStream_t stream) {
  (void)in_sizes; (void)n_in; (void)out_size; (void)ws_size;
  const float* h  = (const float*)d_in[0];   // [400,256]
  const float* ef = (const float*)d_in[1];   // [160000,5]
  const float* W  = (const float*)d_in[2];   // [256,128]
  const float* a1 = (const float*)d_in[3];   // [261,400]
  const float* a3 = (const float*)d_in[4];   // [160000,400]
  float* out = (float*)d_out;                // [400,128]

  float* ws     = (float*)d_ws;
  float* wh     = ws;                    //    51,200
  float* S1     = wh + 51200;            //   160,000
  float* S2     = S1 + 160000;           //   160,000
  float* B1     = S2 + 160000;           //   160,000  [i][r]
  float* B2     = B1 + 160000;           //   160,000  [j][r]
  float* C      = B2 + 160000;           //     2,000
  float* B1part = C + 2000;              // 3,200,000  (GJ x 400 x 400)
  float* B2part = B1part + 3200000;      // 3,200,000  (GI x 400 x 400)
  float* Cpart  = B2part + 3200000;      //   800,000  (400 blk x 2000)
  float* ebuf   = Cpart + 800000;        //   160,000   -> ~32.2 MB total

  // Long pole: ONE non-temporal pass over a3 (256 MB) + ~58 MB partials
  // ~= 314 MB @ 23.3 TB/s ~= 13.5 us.
  k_reduce_a3<<<GI * GJ, 256, 0, stream>>>(a3, ef, B1part, B2part, Cpart);
  k_reduce_parts<<<(2 * NN * NN + NN * 5 + 255) / 256, 256, 0, stream>>>(
      B1part, B2part, Cpart, B1, B2, C);

  // Small WMMA GEMM chain (~0.4 GFLOP total)
  k_gemm_wh<<<200, 32, 0, stream>>>(h, W, wh);
  k_gemm_s<<<1250, 32, 0, stream>>>(wh, a1, S1, S2);
  k_gemm_e<<<625, 32, 0, stream>>>(B1, S1, B2, S2, C, a1, ebuf);
  k_softmax<<<400, 256, 0, stream>>>(ebuf);
  k_gemm_out<<<200, 32, 0, stream>>>(ebuf, wh, out);
}